// SORDefense_9285719294371
// MI455X (gfx1250) — compile-verified
//
#include <hip/hip_runtime.h>
#include <math.h>

#define KPTS   2048
#define BATCH  16
#define NPT    2048
#define ALPHA  1.1f

typedef __attribute__((ext_vector_type(2))) float v2f;
typedef __attribute__((ext_vector_type(8))) float v8f;

// ---------------- Kernel 1: kNN mean distance via WMMA f32 16x16x4 ----------------
// 256-thread block (8 waves) per (batch, group of 8 row-tiles). The batch's
// points (24KB) + squared norms (8KB) are staged in LDS once; each wave then
// runs 128 column tiles: D = (-2A) x B + (si+sj) == distance tile directly.
__global__ __launch_bounds__(256)
void sor_knn_kernel(const float* __restrict__ x, float* __restrict__ value) {
    __shared__ float s_x[3 * KPTS];   // 24 KB, layout [3][K] like input
    __shared__ float s_sq[KPTS];      //  8 KB

    const int t     = threadIdx.x;
    const int b     = blockIdx.x >> 4;   // 16 blocks per batch
    const int grp   = blockIdx.x & 15;   // which group of 8 row-tiles
    const int wid   = t >> 5;            // wave in block (0..7)
    const int lane  = t & 31;
    const int h     = lane >> 4;         // lane half = K msb of A/B fragments
    const int m     = lane & 15;         // row (A) / col (B,C) within tile
    const int ibase = (grp * 8 + wid) * 16;

    const float* xb = x + (size_t)b * 3 * KPTS;

    // Stage batch data in LDS, derive squared norms in LDS (no sq kernel needed).
    for (int i = t; i < 3 * KPTS; i += 256) s_x[i] = xb[i];
    __syncthreads();
    for (int i = t; i < KPTS; i += 256) {
        float x0 = s_x[i], x1 = s_x[KPTS + i], x2 = s_x[2 * KPTS + i];
        s_sq[i] = x0 * x0 + x1 * x1 + x2 * x2;
    }
    __syncthreads();

    // Branchless lane constants: half h==1 supplies K={2,3} => comp z then zero.
    const float ymask = h ? 0.0f : 1.0f;
    const int   zoff  = (h ? 2 : 0) * KPTS;

    // A fragment (ISA 32-bit A 16x4: VGPR0->K=2h, VGPR1->K=2h+1), pre-scaled by -2.
    v2f a;
    a.x = -2.0f * s_x[zoff + ibase + m];
    a.y = -2.0f * (ymask * s_x[KPTS + ibase + m]);

    // Row squared norms per C/D slot: VGPR v holds row M = v + 8*h.
    float si[8];
#pragma unroll
    for (int v = 0; v < 8; ++v) si[v] = s_sq[ibase + v + 8 * h];

    float t0[8], t1[8], t2[8];
#pragma unroll
    for (int v = 0; v < 8; ++v) { t0[v] = 3.0e38f; t1[v] = 3.0e38f; t2[v] = 3.0e38f; }

#pragma unroll 4
    for (int jt = 0; jt < 128; ++jt) {
        const int jbase = jt * 16;
        // B fragment 4x16 from LDS: lanes index N=m, VGPR0->K=2h, VGPR1->K=2h+1.
        v2f bf;
        bf.x = s_x[zoff + jbase + m];
        bf.y = ymask * s_x[KPTS + jbase + m];
        const float sj = s_sq[jbase + m];

        v8f c;
#pragma unroll
        for (int v = 0; v < 8; ++v) c[v] = si[v] + sj;

        c = __builtin_amdgcn_wmma_f32_16x16x4_f32(
                /*neg_a=*/false, a, /*neg_b=*/false, bf,
                /*c_mod=*/(short)0, c, /*reuse_a=*/false, /*reuse_b=*/false);

#pragma unroll
        for (int v = 0; v < 8; ++v) {
            float d  = c[v];
            float n2 = fminf(t2[v], fmaxf(d, t1[v]));
            float n1 = fminf(t1[v], fmaxf(d, t0[v]));
            float n0 = fminf(t0[v], d);
            t0[v] = n0; t1[v] = n1; t2[v] = n2;
        }
    }

    // Butterfly-merge triples across the 16 lanes of each half (xor stays in-half).
#pragma unroll
    for (int v = 0; v < 8; ++v) {
#pragma unroll
        for (int st = 0; st < 4; ++st) {
            const int msk = 1 << st;
            float b0 = __shfl_xor(t0[v], msk, 32);
            float b1 = __shfl_xor(t1[v], msk, 32);
            float b2 = __shfl_xor(t2[v], msk, 32);
            float mn1  = fminf(t1[v], b1);
            float r0   = fminf(t0[v], b0);
            float tmax = fmaxf(t0[v], b0);
            float r1   = fminf(tmax, mn1);
            float r2   = fminf(fmaxf(mn1, tmax), fminf(t2[v], b2));
            t0[v] = r0; t1[v] = r1; t2[v] = r2;
        }
    }

    // value = mean of 2nd+3rd smallest (drop self=min). One writer per row:
    // lanes with bit3==0 write slot s=lane&7 -> row = s + 8*h.
    float vout = 0.0f;
#pragma unroll
    for (int v = 0; v < 8; ++v)
        if ((lane & 7) == v) vout = 0.5f * (t1[v] + t2[v]);
    if (!(lane & 8)) {
        const int row = (lane & 7) + 8 * h;
        value[(size_t)b * KPTS + ibase + row] = vout;
    }
}

// ---------------- Kernel 2: stats + stable compaction + tiled gather ----------------
__global__ __launch_bounds__(256)
void sor_select_kernel(const float* __restrict__ x, const float* __restrict__ value,
                       float* __restrict__ out) {
    __shared__ float s_red[256];
    __shared__ int   s_scan[256];
    __shared__ int   s_kept[KPTS];

    const int b = blockIdx.x;
    const int t = threadIdx.x;
    const float* vb = value + (size_t)b * KPTS;
    const float* xb = x + (size_t)b * 3 * KPTS;
    float* ob = out + (size_t)b * 3 * NPT;

    // Each thread owns 8 consecutive values (stable order for compaction).
    float vv[8];
    float psum = 0.f;
#pragma unroll
    for (int u = 0; u < 8; ++u) { vv[u] = vb[t * 8 + u]; psum += vv[u]; }

    s_red[t] = psum; __syncthreads();
    for (int off = 128; off > 0; off >>= 1) { if (t < off) s_red[t] += s_red[t + off]; __syncthreads(); }
    const float mean = s_red[0] * (1.0f / KPTS);
    __syncthreads();

    float pss = 0.f;
#pragma unroll
    for (int u = 0; u < 8; ++u) { float d = vv[u] - mean; pss += d * d; }
    s_red[t] = pss; __syncthreads();
    for (int off = 128; off > 0; off >>= 1) { if (t < off) s_red[t] += s_red[t + off]; __syncthreads(); }
    const float stdv = sqrtf(s_red[0] / (float)(KPTS - 1));   // ddof=1
    const float thr  = mean + ALPHA * stdv;

    int cnt = 0;
#pragma unroll
    for (int u = 0; u < 8; ++u) cnt += (vv[u] <= thr) ? 1 : 0;

    __syncthreads();
    s_scan[t] = cnt; __syncthreads();
    for (int off = 1; off < 256; off <<= 1) {
        int mine = s_scan[t];
        int add  = (t >= off) ? s_scan[t - off] : 0;
        __syncthreads();
        s_scan[t] = mine + add;
        __syncthreads();
    }
    int n   = s_scan[255];
    int pos = s_scan[t] - cnt;        // exclusive prefix
#pragma unroll
    for (int u = 0; u < 8; ++u)
        if (vv[u] <= thr) s_kept[pos++] = t * 8 + u;
    __syncthreads();

    if (n < 1) n = 1;                 // always >=1 in theory; guard anyway
    for (int j = t; j < NPT; j += 256) {
        int idx = s_kept[j % n] & (KPTS - 1);   // clamp: no-op when valid
        ob[j]           = xb[idx];
        ob[NPT + j]     = xb[KPTS + idx];
        ob[2 * NPT + j] = xb[2 * KPTS + idx];
    }
}

extern "C" void kernel_launch(void* const* d_in, const int* in_sizes, int n_in,
                              void* d_out, int out_size, void* d_ws, size_t ws_size,
                              hipStream_t stream) {
    (void)in_sizes; (void)n_in; (void)out_size; (void)ws_size;
    const float* x = (const float*)d_in[0];     // [B,3,K] fp32
    float* out     = (float*)d_out;             // [B,3,NPT] fp32
    float* value   = (float*)d_ws;              // B*K floats scratch

    sor_knn_kernel   <<<BATCH * 16, 256, 0, stream>>>(x, value);
    sor_select_kernel<<<BATCH,      256, 0, stream>>>(x, value, out);
}